// SelectiveSSM_37958920962158
// MI455X (gfx1250) — compile-verified
//
#include <hip/hip_runtime.h>
#include <math.h>

// ---------------------------------------------------------------------------
// SelectiveSSM for MI455X (gfx1250, wave32)
//   B=4, L=2048, DM=1024, DS=16, DTR=64
//   Phase 1: proj = x @ Wx^T          f32 WMMA 16x16x4  (M=8192,K=1024,N=96)
//   Phase 2: dt = softplus(dt_r@dt_W^T + b)  f32 WMMA   (M=8192,K=64,N=1024)
//   Phase 3: diagonal selective scan, one lane per (d,s) recurrence,
//            shfl_xor butterfly for the s-reduction.
// ---------------------------------------------------------------------------

typedef __attribute__((ext_vector_type(2))) float v2f;
typedef __attribute__((ext_vector_type(8))) float v8f;

#define BATCH   4
#define SEQLEN  2048
#define DMODEL  1024
#define DSTATE  16
#define DTRANK  64
#define NPROJ   96            // dt_rank + 2*d_state
#define MROWS   (BATCH*SEQLEN) // 8192

// ---------------------------------------------------------------------------
// GEMM1: proj[bl,p] = sum_k x[bl,k] * Wx[p,k]
// One wave per 16-row M tile; each wave produces all 6 N-tiles (96 cols),
// reusing the A fragment 6x per K step. 512 waves total.
// ---------------------------------------------------------------------------
__global__ void ssm_proj_gemm(const float* __restrict__ x,
                              const float* __restrict__ Wx,
                              float* __restrict__ proj)
{
    const int wave = (blockIdx.x * blockDim.x + threadIdx.x) >> 5;
    const int lane = threadIdx.x & 31;
    const int lrow = lane & 15;
    const int hi   = lane >> 4;          // 0: K=0,1   1: K=2,3
    const int row0 = wave * 16;          // grid sized exactly: no guard needed

    v8f acc[6];
    const v8f vzero = {0.f,0.f,0.f,0.f,0.f,0.f,0.f,0.f};
#pragma unroll
    for (int j = 0; j < 6; ++j) acc[j] = vzero;

    const float* xrow = x + (size_t)(row0 + lrow) * DMODEL;

    for (int k0 = 0; k0 < DMODEL; k0 += 4) {
        // A fragment: 16x4 tile of x
        const v2f a = *(const v2f*)(xrow + k0 + 2*hi);
#pragma unroll
        for (int j = 0; j < 6; ++j) {
            // B fragment: B[k][n] = Wx[j*16+n][k0+k]
            const v2f b = *(const v2f*)(Wx + (size_t)(j*16 + lrow) * DMODEL + k0 + 2*hi);
            acc[j] = __builtin_amdgcn_wmma_f32_16x16x4_f32(
                false, a, false, b, (short)0, acc[j], false, false);
        }
    }

#pragma unroll
    for (int j = 0; j < 6; ++j)
#pragma unroll
        for (int r = 0; r < 8; ++r)
            proj[(size_t)(row0 + r + 8*hi) * NPROJ + j*16 + lrow] = acc[j][r];
}

// ---------------------------------------------------------------------------
// GEMM2: dt[bl,d] = softplus( sum_r proj[bl,r] * dt_W[d,r] + dt_b[d] )
// One wave per (16-row M tile, 128-col N group): 512*8 = 4096 waves.
// ---------------------------------------------------------------------------
__global__ void ssm_dt_gemm(const float* __restrict__ proj,
                            const float* __restrict__ dtW,
                            const float* __restrict__ dtb,
                            float* __restrict__ dt)
{
    const int wave  = (blockIdx.x * blockDim.x + threadIdx.x) >> 5;
    const int lane  = threadIdx.x & 31;
    const int lrow  = lane & 15;
    const int hi    = lane >> 4;
    const int mtile = wave >> 3;         // 0..511
    const int ngrp  = wave & 7;          // 0..7  -> cols [ngrp*128, +128)
    const int row0  = mtile * 16;
    const int col0  = ngrp * 128;

    v8f acc[8];
    const v8f vzero = {0.f,0.f,0.f,0.f,0.f,0.f,0.f,0.f};
#pragma unroll
    for (int j = 0; j < 8; ++j) acc[j] = vzero;

    const float* arow = proj + (size_t)(row0 + lrow) * NPROJ;   // dt_r = proj[:, 0:64]

#pragma unroll
    for (int k0 = 0; k0 < DTRANK; k0 += 4) {
        const v2f a = *(const v2f*)(arow + k0 + 2*hi);
#pragma unroll
        for (int j = 0; j < 8; ++j) {
            const v2f b = *(const v2f*)(dtW + (size_t)(col0 + j*16 + lrow) * DTRANK + k0 + 2*hi);
            acc[j] = __builtin_amdgcn_wmma_f32_16x16x4_f32(
                false, a, false, b, (short)0, acc[j], false, false);
        }
    }

#pragma unroll
    for (int j = 0; j < 8; ++j) {
        const int col  = col0 + j*16 + lrow;
        const float bias = dtb[col];
#pragma unroll
        for (int r = 0; r < 8; ++r) {
            float z  = acc[j][r] + bias;
            float sp = (z > 20.0f) ? z : log1pf(__expf(z));   // softplus
            dt[(size_t)(row0 + r + 8*hi) * DMODEL + col] = sp;
        }
    }
}

// ---------------------------------------------------------------------------
// Phase 3: scan. One lane per (d,s) scalar recurrence:
//   h = exp(dt*A[d,s]) * h + (dt*x) * B_t[s];   y[d] = sum_s h * C_t[s]
// Block = 256 threads = 16 d-channels x 16 states, one batch per blockIdx.y.
// s = tid&15 so the shfl_xor {1,2,4,8} butterfly reduces within a d-group.
// 65536 lanes = 2048 waves of parallel work.
// ---------------------------------------------------------------------------
__global__ void ssm_scan(const float* __restrict__ x,
                         const float* __restrict__ A_log,
                         const float* __restrict__ Dp,
                         const float* __restrict__ proj,
                         const float* __restrict__ dt,
                         float* __restrict__ y)
{
    const int s = threadIdx.x & 15;
    const int d = blockIdx.x * 16 + (threadIdx.x >> 4);
    const int b = blockIdx.y;

    const float A  = -__expf(A_log[(size_t)d * DSTATE + s]);
    const float Dd = Dp[d];
    float h = 0.0f;

    const size_t bl0 = (size_t)b * SEQLEN;

    for (int l = 0; l < SEQLEN; ++l) {
        const size_t bl = bl0 + l;
        const float dtv = dt[bl * DMODEL + d];
        const float xv  = x [bl * DMODEL + d];
        const float* prow = proj + bl * NPROJ;   // uniform per block -> scalar loads
        const float Bv = prow[DTRANK + s];
        const float Cv = prow[DTRANK + DSTATE + s];

        const float dA = __expf(dtv * A);
        h = fmaf(h, dA, dtv * xv * Bv);

        float part = h * Cv;
        part += __shfl_xor(part, 1);
        part += __shfl_xor(part, 2);
        part += __shfl_xor(part, 4);
        part += __shfl_xor(part, 8);     // every lane now holds sum over its 16-group

        if (s == 0)
            y[bl * DMODEL + d] = part + xv * Dd;
    }
}

// ---------------------------------------------------------------------------
extern "C" void kernel_launch(void* const* d_in, const int* in_sizes, int n_in,
                              void* d_out, int out_size, void* d_ws, size_t ws_size,
                              hipStream_t stream)
{
    const float* x    = (const float*)d_in[0];   // (4,2048,1024)
    const float* Alog = (const float*)d_in[1];   // (1024,16)
    const float* D    = (const float*)d_in[2];   // (1024)
    const float* Wx   = (const float*)d_in[3];   // (96,1024)
    const float* dtW  = (const float*)d_in[4];   // (1024,64)
    const float* dtb  = (const float*)d_in[5];   // (1024)
    float* out = (float*)d_out;                  // (4,2048,1024)

    float* proj = (float*)d_ws;                  // 8192*96   floats (3 MB)
    float* dt   = proj + (size_t)MROWS * NPROJ;  // 8192*1024 floats (32 MB)

    // GEMM1: 512 M-tiles, 1 wave each -> 64 blocks x 8 waves (exact, no guards)
    ssm_proj_gemm<<<64, 256, 0, stream>>>(x, Wx, proj);

    // GEMM2: 512 M-tiles x 8 N-groups = 4096 waves -> 512 blocks x 8 waves
    ssm_dt_gemm<<<512, 256, 0, stream>>>(proj, dtW, dtb, dt);

    // Scan: grid (64 d-groups, 4 batches), 256 threads (16 d x 16 s)
    ssm_scan<<<dim3(64, BATCH), 256, 0, stream>>>(x, Alog, D, proj, dt, out);
}